// SINDy_SHRED_47485158424896
// MI455X (gfx1250) — compile-verified
//
#include <hip/hip_runtime.h>

#define LATENT_DIM 16
#define NUM_REPLICATES 10
#define NUM_STEPS 10
#define NUM_TERMS 169          // 1 + 16 + 136 + 16
#define KCHUNKS 43             // ceil(169/4) -> K padded to 172
#define DT 0.01f
#define TILES_PER_BLOCK 5
#define NTILES 3125            // 50000 / 16
#define NBLOCKS (NTILES / TILES_PER_BLOCK)  // 625
#define ROWPAD 20              // LDS row stride in floats (80B: 16B aligned, bank-conflict free)
#define INV_2PI 0.15915494309189535f

typedef __attribute__((ext_vector_type(2))) float v2f;
typedef __attribute__((ext_vector_type(8))) float v8f;

// 16 latent values are carried as individual SSA floats -- never as an
// indexable array -- so the backend can never materialize v_movrels.
#define ZPARAMS float z0, float z1, float z2, float z3, float z4, float z5,  \
                float z6, float z7, float z8, float z9, float z10, float z11,\
                float z12, float z13, float z14, float z15
#define ZARGS z0, z1, z2, z3, z4, z5, z6, z7, z8, z9, z10, z11, z12, z13, z14, z15

template <int I>
__device__ __forceinline__ float pick(ZPARAMS) {
    if constexpr (I == 0)  return z0;
    else if constexpr (I == 1)  return z1;
    else if constexpr (I == 2)  return z2;
    else if constexpr (I == 3)  return z3;
    else if constexpr (I == 4)  return z4;
    else if constexpr (I == 5)  return z5;
    else if constexpr (I == 6)  return z6;
    else if constexpr (I == 7)  return z7;
    else if constexpr (I == 8)  return z8;
    else if constexpr (I == 9)  return z9;
    else if constexpr (I == 10) return z10;
    else if constexpr (I == 11) return z11;
    else if constexpr (I == 12) return z12;
    else if constexpr (I == 13) return z13;
    else if constexpr (I == 14) return z14;
    else return z15;
}

// triu_indices(16) term -> (i, j), standard SINDy row-major ordering
constexpr int tri_i(int u) {
    int i = 0, rem = u;
    while (rem >= LATENT_DIM - i) { rem -= LATENT_DIM - i; ++i; }
    return i;
}
constexpr int tri_j(int u) {
    int i = 0, rem = u;
    while (rem >= LATENT_DIM - i) { rem -= LATENT_DIM - i; ++i; }
    return i + rem;
}

// Library term T (compile-time index). sin via hardware v_sin_f32
// (computes sin(2*pi*x)) -> pre-scale by 1/(2*pi).
template <int T>
__device__ __forceinline__ float term(ZPARAMS) {
    if constexpr (T == 0) {
        return 1.0f;
    } else if constexpr (T >= 1 && T <= 16) {
        return pick<T - 1>(ZARGS);
    } else if constexpr (T >= 17 && T <= 152) {
        return pick<tri_i(T - 17)>(ZARGS) * pick<tri_j(T - 17)>(ZARGS);
    } else if constexpr (T >= 153 && T <= 168) {
        return __builtin_amdgcn_sinf(pick<T - 153>(ZARGS) * INV_2PI);
    } else {
        return 0.0f;  // K padding 169..171
    }
}

// Fully-unrolled K loop: 43 x V_WMMA_F32_16X16X4_F32, accumulating into acc (= h state).
// Lane-half `hi` selects which 86 library terms this lane supplies (free K permutation:
// library index t = 86*hi + 2*C + s  <->  WMMA K = 4*C + 2*hi + s).
template <int C>
struct ChunkLoop {
    static __device__ __forceinline__ void run(v8f& acc, const v2f (&B)[KCHUNKS],
                                               int hi, ZPARAMS) {
        const float lo0 = term<2 * C>(ZARGS);
        const float hi0 = term<86 + 2 * C>(ZARGS);
        const float lo1 = term<2 * C + 1>(ZARGS);
        const float hi1 = term<86 + 2 * C + 1>(ZARGS);
        v2f A;
        A.x = hi ? hi0 : lo0;
        A.y = hi ? hi1 : lo1;
        acc = __builtin_amdgcn_wmma_f32_16x16x4_f32(
            /*neg_a=*/false, A, /*neg_b=*/false, B[C],
            /*c_mod=*/(short)0, acc, /*reuse_a=*/false, /*reuse_b=*/false);
        ChunkLoop<C + 1>::run(acc, B, hi, ZARGS);
    }
};
template <>
struct ChunkLoop<KCHUNKS> {
    static __device__ __forceinline__ void run(v8f&, const v2f (&)[KCHUNKS], int, ZPARAMS) {}
};

__global__ __launch_bounds__(32 * NUM_REPLICATES)
void sindy_shred_wmma_kernel(const float* __restrict__ h_t,
                             const float* __restrict__ coeff,
                             const float* __restrict__ cmask,
                             float* __restrict__ out) {
    __shared__ float lds[NUM_REPLICATES][LATENT_DIM * ROWPAD];

    const int lane = threadIdx.x & 31;
    const int r    = threadIdx.x >> 5;   // wave index == replicate
    const int hi   = lane >> 4;          // lane half (supplies K rows 2,3 of each chunk)
    const int lo   = lane & 15;
    float* myLds   = &lds[r][0];

    // ---- Load B operand once per wave: C_r * mask * DT, permuted into WMMA B layout.
    // Lane supplies column n=lo for library rows t = 86*hi + 2c + {0,1}.
    v2f B[KCHUNKS];
    {
        const float* cf = coeff + (size_t)r * NUM_TERMS * LATENT_DIM + lo;
        const float* mk = cmask + (size_t)r * NUM_TERMS * LATENT_DIM + lo;
        const int tb = hi * 86;
#pragma unroll
        for (int c = 0; c < KCHUNKS; ++c) {
            const int t0 = tb + 2 * c;
            const int t1 = t0 + 1;
            float b0 = (t0 < NUM_TERMS) ? cf[t0 * LATENT_DIM] * mk[t0 * LATENT_DIM] * DT : 0.0f;
            float b1 = (t1 < NUM_TERMS) ? cf[t1 * LATENT_DIM] * mk[t1 * LATENT_DIM] * DT : 0.0f;
            v2f b; b.x = b0; b.y = b1;
            B[c] = b;
        }
    }

    for (int tIdx = 0; tIdx < TILES_PER_BLOCK; ++tIdx) {
        const int n0 = (blockIdx.x * TILES_PER_BLOCK + tIdx) * 16;

        // h state lives in the 16x16 f32 C/D register layout:
        // acc[v] = h[n0 + v + 8*hi][lo]
        v8f acc;
#pragma unroll
        for (int v = 0; v < 8; ++v)
            acc[v] = h_t[(size_t)(n0 + v + 8 * hi) * LATENT_DIM + lo];

        for (int step = 0; step < NUM_STEPS; ++step) {
            // Transpose h tile through per-wave LDS so each lane gets its full row z[16].
#pragma unroll
            for (int v = 0; v < 8; ++v)
                myLds[(v + 8 * hi) * ROWPAD + lo] = acc[v];
            asm volatile("s_wait_dscnt 0" ::: "memory");

            const float4 t0 = *(const float4*)&myLds[lo * ROWPAD + 0];
            const float4 t1 = *(const float4*)&myLds[lo * ROWPAD + 4];
            const float4 t2 = *(const float4*)&myLds[lo * ROWPAD + 8];
            const float4 t3 = *(const float4*)&myLds[lo * ROWPAD + 12];

            // h += Theta(h) @ (C_r*mask*DT): 43 chained f32 WMMAs, D->C accumulation.
            ChunkLoop<0>::run(acc, B, hi,
                              t0.x, t0.y, t0.z, t0.w,
                              t1.x, t1.y, t1.z, t1.w,
                              t2.x, t2.y, t2.z, t2.w,
                              t3.x, t3.y, t3.z, t3.w);
        }

        // out[n][r][d]
#pragma unroll
        for (int v = 0; v < 8; ++v)
            out[((size_t)(n0 + v + 8 * hi) * NUM_REPLICATES + r) * LATENT_DIM + lo] = acc[v];
    }
}

extern "C" void kernel_launch(void* const* d_in, const int* in_sizes, int n_in,
                              void* d_out, int out_size, void* d_ws, size_t ws_size,
                              hipStream_t stream) {
    (void)in_sizes; (void)n_in; (void)d_ws; (void)ws_size; (void)out_size;
    const float* h_t   = (const float*)d_in[0];
    const float* coeff = (const float*)d_in[1];
    const float* cmask = (const float*)d_in[2];
    float* out = (float*)d_out;

    sindy_shred_wmma_kernel<<<NBLOCKS, 32 * NUM_REPLICATES, 0, stream>>>(h_t, coeff, cmask, out);
}